// EdgeAwareCritic_87797721465078
// MI455X (gfx1250) — compile-verified
//
#include <hip/hip_runtime.h>

// ---------------------------------------------------------------------------
// EdgeAwareCritic for MI455X (gfx1250, wave32, WMMA).
// Dense node GEMMs use V_WMMA_F32_16X16X4_F32 (full fp32 precision).
// W tiles are staged through LDS and shared by 4 waves per block; the K loop
// is a compile-time constant so it fully unrolls and the A-stream global
// loads can be pipelined ahead of the WMMA waits.
// Edge message GEMV (K=16) is fused into the edge kernel to avoid
// materializing the 327MB [E,256] tensor (keeps everything L2-resident).
// ---------------------------------------------------------------------------

typedef __attribute__((ext_vector_type(2))) float v2f;
typedef __attribute__((ext_vector_type(8))) float v8f;

#define N_NODES 20000
#define N_EDGES 320000
#define IN_DIM  128
#define HID     128
#define ADIM    16
#define EDIM    16
#define NHEADS  2

// ---- order-preserving float <-> uint encoding for atomicMax on floats ------
__device__ __forceinline__ unsigned f2ord(float f) {
    unsigned u = __float_as_uint(f);
    return (u & 0x80000000u) ? ~u : (u | 0x80000000u);
}
__device__ __forceinline__ float ord2f(unsigned u) {
    return __uint_as_float((u & 0x80000000u) ? (u ^ 0x80000000u) : ~u);
}

// ---------------------------------------------------------------------------
// GEMM: Y[rows, M] = act(X[rows, K] @ W[M, K]^T + bias)
// Block = 128 threads = 4 waves. All 4 waves share one 64-wide column slab of
// W staged in LDS; each wave owns a distinct 16-row tile. Per wave: 4 WMMA
// f32 16x16x4 accumulators (16 rows x 64 cols).
// WMMA f32 16x16x4 layout (wave32):
//   A (16x4):  lanes 0-15 hold M=lane, {K0,K1}; lanes 16-31 hold M=lane-16, {K2,K3}
//   B (4x16):  lanes 0-15 hold N=lane, {K0,K1}; lanes 16-31 hold N=lane-16, {K2,K3}
//   C/D(16x16):lane<16 -> col=lane, row=v;  lane>=16 -> col=lane-16, row=v+8
// ---------------------------------------------------------------------------
template<int K, int RELU>
__global__ void gemm_wmma_k(const float* __restrict__ X,
                            const float* __restrict__ W,
                            const float* __restrict__ bias,
                            float* __restrict__ Y,
                            int rows, int M)
{
    __shared__ float Bs[64 * K];               // 32KB for K=128 (WGP has 320KB)

    const int tid  = threadIdx.x;              // 0..127
    const int wave = tid >> 5;                 // 0..3
    const int lane = tid & 31;
    const int half = lane >> 4;                // which K pair / row half
    const int l16  = lane & 15;
    const int col0 = blockIdx.y << 6;

    // Cooperatively stage W[col0..col0+63][0..K) into LDS, row-major [64][K].
    constexpr int NV4 = 64 * K / 4;
#pragma unroll
    for (int idx = tid; idx < NV4; idx += 128) {
        const int c  = idx / (K / 4);
        const int kk = (idx - c * (K / 4)) * 4;
        *(float4*)&Bs[c * K + kk] = *(const float4*)&W[(size_t)(col0 + c) * K + kk];
    }
    __syncthreads();

    // Each wave takes its own 16-row tile; clamp trailing waves so EXEC stays
    // all-ones for WMMA (duplicate same-value stores are benign).
    int row0 = (blockIdx.x * 4 + wave) << 4;
    if (row0 > rows - 16) row0 = rows - 16;

    v8f acc[4];
#pragma unroll
    for (int t = 0; t < 4; ++t) acc[t] = (v8f){0.f,0.f,0.f,0.f,0.f,0.f,0.f,0.f};

    const float* xrow = X + (size_t)(row0 + l16) * K + 2 * half;
    const float* bs   = Bs + l16 * K + 2 * half;

#pragma unroll
    for (int k = 0; k < K; k += 4) {
        v2f a  = *(const v2f*)(xrow + k);
        v2f b0 = *(const v2f*)(bs + 0 * 16 * K + k);
        v2f b1 = *(const v2f*)(bs + 1 * 16 * K + k);
        v2f b2 = *(const v2f*)(bs + 2 * 16 * K + k);
        v2f b3 = *(const v2f*)(bs + 3 * 16 * K + k);
        acc[0] = __builtin_amdgcn_wmma_f32_16x16x4_f32(false, a, false, b0, (short)0, acc[0], false, false);
        acc[1] = __builtin_amdgcn_wmma_f32_16x16x4_f32(false, a, false, b1, (short)0, acc[1], false, false);
        acc[2] = __builtin_amdgcn_wmma_f32_16x16x4_f32(false, a, false, b2, (short)0, acc[2], false, false);
        acc[3] = __builtin_amdgcn_wmma_f32_16x16x4_f32(false, a, false, b3, (short)0, acc[3], false, false);
    }

#pragma unroll
    for (int t = 0; t < 4; ++t) {
        const int col = col0 + t * 16 + l16;
        const float bv = bias[col];
#pragma unroll
        for (int v = 0; v < 8; ++v) {
            float val = acc[t][v] + bv;
            if (RELU) val = fmaxf(val, 0.f);
            Y[(size_t)(row0 + v + 8 * half) * M + col] = val;
        }
    }
}

static void launch_gemm(const float* X, const float* W, const float* b, float* Y,
                        int K, int rows, int M, int relu, hipStream_t s)
{
    dim3 grid((rows / 16 + 3) / 4, M / 64);
    dim3 blk(128);
    if (K == 128) {
        if (relu) gemm_wmma_k<128, 1><<<grid, blk, 0, s>>>(X, W, b, Y, rows, M);
        else      gemm_wmma_k<128, 0><<<grid, blk, 0, s>>>(X, W, b, Y, rows, M);
    } else {
        if (relu) gemm_wmma_k<16, 1><<<grid, blk, 0, s>>>(X, W, b, Y, rows, M);
        else      gemm_wmma_k<16, 0><<<grid, blk, 0, s>>>(X, W, b, Y, rows, M);
    }
}

// ---------------------------------------------------------------------------
// Edge phase A: one wave per edge. Fuses em = ea @ We^T (K=16) with the
// leaky-relu message and the attention dot. Lanes 0-15 -> head 0 channels,
// lanes 16-31 -> head 1 channels (8 channels each). Writes raw logit and
// atomically tracks per-(dst,head) max for a stable segment softmax.
// ---------------------------------------------------------------------------
__global__ void edge_logit(const int* __restrict__ src, const int* __restrict__ dst,
                           const float* __restrict__ ea,   // [E,16]
                           const float* __restrict__ We,   // [256,16]
                           const float* __restrict__ att,  // [256]
                           const float* __restrict__ xl,   // [N,256]
                           const float* __restrict__ xr,   // [N,256]
                           float* __restrict__ wbuf,       // [E,2] raw logits
                           unsigned* __restrict__ lmax,    // [N,2] ordered-enc max
                           int E)
{
    const int gid  = blockIdx.x * blockDim.x + threadIdx.x;
    const int e    = gid >> 5;
    if (e >= E) return;
    const int lane  = gid & 31;
    const int l16   = lane & 15;
    const int h     = lane >> 4;
    const int cbase = lane * 8;           // channel base in [0,256)

    const int s = src[e], d = dst[e];

    float eav[16];
#pragma unroll
    for (int j = 0; j < 16; ++j) eav[j] = ea[(size_t)e * 16 + j];

    const float* xls = xl + (size_t)s * 256 + cbase;
    const float* xrd = xr + (size_t)d * 256 + cbase;

    float dot = 0.f;
#pragma unroll
    for (int c = 0; c < 8; ++c) {
        const int ch = cbase + c;
        const float* wr = We + (size_t)ch * 16;
        float em = 0.f;
#pragma unroll
        for (int j = 0; j < 16; ++j) em = fmaf(eav[j], wr[j], em);
        float m = xls[c] + xrd[c] + em;
        m = (m > 0.f) ? m : 0.2f * m;                 // leaky_relu(0.2)
        dot = fmaf(m, att[ch], dot);
    }
    // reduce across the 16-lane half (stays within one head)
    dot += __shfl_xor(dot, 8, 32);
    dot += __shfl_xor(dot, 4, 32);
    dot += __shfl_xor(dot, 2, 32);
    dot += __shfl_xor(dot, 1, 32);

    if (l16 == 0) {
        wbuf[(size_t)e * 2 + h] = dot;
        atomicMax(&lmax[(size_t)d * 2 + h], f2ord(dot));
    }
}

// Edge phase B: w = exp(logit - max[dst]); accumulate denominator.
__global__ void softmax_w(float* __restrict__ wbuf, const int* __restrict__ dst,
                          const unsigned* __restrict__ lmax, float* __restrict__ den,
                          int E)
{
    const int i = blockIdx.x * blockDim.x + threadIdx.x;
    if (i >= E * 2) return;
    const int e = i >> 1, h = i & 1;
    const int d = dst[e];
    const float w = __expf(wbuf[i] - ord2f(lmax[(size_t)d * 2 + h]));
    wbuf[i] = w;
    atomicAdd(&den[(size_t)d * 2 + h], w);
}

// Edge phase C: acc[dst] += alpha * xl[src]; one wave per edge, 8 ch/lane.
__global__ void scatter_acc(const float* __restrict__ wbuf, const float* __restrict__ den,
                            const int* __restrict__ src, const int* __restrict__ dst,
                            const float* __restrict__ xl, float* __restrict__ acc, int E)
{
    const int gid = blockIdx.x * blockDim.x + threadIdx.x;
    const int e   = gid >> 5;
    if (e >= E) return;
    const int lane  = gid & 31;
    const int h     = lane >> 4;
    const int cbase = lane * 8;
    const int s = src[e], d = dst[e];
    const float alpha = wbuf[(size_t)e * 2 + h] / den[(size_t)d * 2 + h];
    const float* xs = xl + (size_t)s * 256 + cbase;
    float* ad = acc + (size_t)d * 256 + cbase;
#pragma unroll
    for (int c = 0; c < 8; ++c) atomicAdd(&ad[c], alpha * xs[c]);
}

// Phase D: mean over heads + bias + relu  -> h_out [N,128]
__global__ void head_mean_bias_relu(const float* __restrict__ acc,
                                    const float* __restrict__ bias,
                                    float* __restrict__ out, int n)
{
    const int i = blockIdx.x * blockDim.x + threadIdx.x;
    if (i >= n) return;
    const int nd = i >> 7, c = i & 127;
    const float v = 0.5f * (acc[(size_t)nd * 256 + c] + acc[(size_t)nd * 256 + 128 + c]) + bias[c];
    out[i] = fmaxf(v, 0.f);
}

// FiLM: out = gamma * h + beta (elementwise)
__global__ void film_apply(const float* __restrict__ g, const float* __restrict__ b,
                           const float* __restrict__ h, float* __restrict__ out, int n)
{
    const int i = blockIdx.x * blockDim.x + threadIdx.x;
    if (i < n) out[i] = fmaf(g[i], h[i], b[i]);
}

__global__ void fill_f32(float* p, float v, int n)
{
    const int i = blockIdx.x * blockDim.x + threadIdx.x;
    if (i < n) p[i] = v;
}
__global__ void fill_u32(unsigned* p, unsigned v, int n)
{
    const int i = blockIdx.x * blockDim.x + threadIdx.x;
    if (i < n) p[i] = v;
}
__global__ void set_scalar(float* out, const float* q2b)
{
    if (threadIdx.x == 0 && blockIdx.x == 0) out[0] = q2b[0];
}

// Final: out[0] += (1/N) * sum_n h[n] . q2w   (block reduce in LDS + atomic)
__global__ void q2_reduce(const float* __restrict__ h, const float* __restrict__ w,
                          float* __restrict__ out, int n)
{
    __shared__ float sm[256];
    const int tid = threadIdx.x;
    const int i   = blockIdx.x * blockDim.x + tid;
    float v = 0.f;
    if (i < n) {
        float dot = 0.f;
#pragma unroll 4
        for (int c = 0; c < 128; ++c) dot = fmaf(h[(size_t)i * 128 + c], w[c], dot);
        v = dot;
    }
    sm[tid] = v;
    __syncthreads();
    for (int s = 128; s > 0; s >>= 1) {
        if (tid < s) sm[tid] += sm[tid + s];
        __syncthreads();
    }
    if (tid == 0) atomicAdd(out, sm[0] * (1.0f / (float)N_NODES));
}

// ---------------------------------------------------------------------------
extern "C" void kernel_launch(void* const* d_in, const int* in_sizes, int n_in,
                              void* d_out, int out_size, void* d_ws, size_t ws_size,
                              hipStream_t stream)
{
    const int N = N_NODES, E = N_EDGES;

    const float* x      = (const float*)d_in[0];
    const int*   ei     = (const int*)d_in[1];
    const float* ea     = (const float*)d_in[2];
    const float* action = (const float*)d_in[3];
    const float* inpW   = (const float*)d_in[4];
    const float* inpb   = (const float*)d_in[5];
    // GAT params: s1 @6..12, s2 @13..19, sa @20..26 (Wl,bl,Wr,br,We,att,bias)
    const float* apW = (const float*)d_in[27];
    const float* apb = (const float*)d_in[28];
    const float* gW  = (const float*)d_in[29];
    const float* gb  = (const float*)d_in[30];
    const float* beW = (const float*)d_in[31];
    const float* beb = (const float*)d_in[32];
    const float* q1W = (const float*)d_in[33];
    const float* q1b = (const float*)d_in[34];
    const float* q2W = (const float*)d_in[35];
    const float* q2b = (const float*)d_in[36];

    const int* srcp = ei;
    const int* dstp = ei + E;

    // workspace layout (bytes); NB = one [N,128] f32 buffer  (total ~85MB)
    const size_t NB = (size_t)N * HID * sizeof(float);
    char* ws = (char*)d_ws;
    float*    hA   = (float*)(ws + 0 * NB);
    float*    hB   = (float*)(ws + 1 * NB);
    float*    xl   = (float*)(ws + 2 * NB);   // [N,256]
    float*    xr   = (float*)(ws + 4 * NB);   // [N,256]
    float*    accb = (float*)(ws + 6 * NB);   // [N,256]
    float*    wbuf = (float*)(ws + 8 * NB);                         // [E,2]
    unsigned* lmax = (unsigned*)(ws + 8 * NB + (size_t)E * 2 * 4);  // [N,2]
    float*    den  = (float*)((char*)lmax + (size_t)N * 2 * 4);     // [N,2]
    float*    aBuf  = xl;               // [N,128] reuse (between GAT layers)
    float*    gamma = xr;               // [N,128] reuse
    float*    beta  = xr + (size_t)N * HID;

    auto run_gat = [&](const float* hin, float* hout, int pbase) {
        const float* Wl   = (const float*)d_in[pbase + 0];
        const float* bl   = (const float*)d_in[pbase + 1];
        const float* Wr   = (const float*)d_in[pbase + 2];
        const float* br   = (const float*)d_in[pbase + 3];
        const float* We   = (const float*)d_in[pbase + 4];
        const float* att  = (const float*)d_in[pbase + 5];
        const float* bias = (const float*)d_in[pbase + 6];

        launch_gemm(hin, Wl, bl, xl, HID, N, 2 * HID, 0, stream);
        launch_gemm(hin, Wr, br, xr, HID, N, 2 * HID, 0, stream);

        fill_u32<<<(N * 2 + 255) / 256, 256, 0, stream>>>(lmax, 0x007FFFFFu, N * 2); // f2ord(-inf)
        fill_f32<<<(N * 2 + 255) / 256, 256, 0, stream>>>(den, 0.f, N * 2);
        fill_f32<<<(N * 256 + 255) / 256, 256, 0, stream>>>(accb, 0.f, N * 256);

        edge_logit<<<E / 8, 256, 0, stream>>>(srcp, dstp, ea, We, att, xl, xr, wbuf, lmax, E);
        softmax_w<<<(E * 2 + 255) / 256, 256, 0, stream>>>(wbuf, dstp, lmax, den, E);
        scatter_acc<<<E / 8, 256, 0, stream>>>(wbuf, den, srcp, dstp, xl, accb, E);
        head_mean_bias_relu<<<(N * HID + 255) / 256, 256, 0, stream>>>(accb, bias, hout, N * HID);
    };

    // h = relu(x @ inp_W.T + inp_b)
    launch_gemm(x, inpW, inpb, hA, IN_DIM, N, HID, 1, stream);

    run_gat(hA, hB, 6);    // s1 (+relu in head_mean)
    run_gat(hB, hA, 13);   // s2 (+relu)

    // FiLM: a = relu(action @ ap_W.T + ap_b); h = (a@gW.T+gb)*h + (a@beW.T+beb)
    launch_gemm(action, apW, apb, aBuf, ADIM, N, HID, 1, stream);
    launch_gemm(aBuf, gW, gb, gamma, HID, N, HID, 0, stream);
    launch_gemm(aBuf, beW, beb, beta, HID, N, HID, 0, stream);
    film_apply<<<(N * HID + 255) / 256, 256, 0, stream>>>(gamma, beta, hA, hB, N * HID);

    run_gat(hB, hA, 20);   // sa (+relu)

    // h = relu(h @ q1_W.T + q1_b)
    launch_gemm(hA, q1W, q1b, hB, HID, N, HID, 1, stream);

    // q = mean_n(h @ q2_W.T + q2_b)
    set_scalar<<<1, 1, 0, stream>>>((float*)d_out, q2b);
    q2_reduce<<<(N + 255) / 256, 256, 0, stream>>>(hB, q2W, (float*)d_out, N);
}